// GKernel_65884798320720
// MI455X (gfx1250) — compile-verified
//
#include <hip/hip_runtime.h>
#include <math.h>

typedef __attribute__((ext_vector_type(2))) float v2f;
typedef __attribute__((ext_vector_type(8))) float v8f;

#define G_     8
#define KS_    7
#define P_     49      // KS*KS
#define KSTEPS 13      // ceil(49/4)
#define KPAD   52      // KSTEPS*4
#define W1STR  54      // padded LDS row stride (words), 8B-aligned rows, conflict-friendly
#define NPAD   64      // 4 N-tiles of 16
#define CINBLK 64
#define ROWS_  512     // CINBLK * G_
#define THREADS 512

__global__ __launch_bounds__(THREADS)
void gconv_expand_wmma(const float* __restrict__ weight,   // [128,128,8,7,7]
                       const float* __restrict__ in_H,     // [8]
                       const float* __restrict__ out_H,    // [8]
                       const float* __restrict__ grid_Rn,  // [7,7,2]
                       const float* __restrict__ mask,     // [7,7]
                       float* __restrict__ out)            // [128,8,128,8,7,7]
{
    __shared__ float sW1[ROWS_ * W1STR];     // 110,592 B : H-interpolated weight tile
    __shared__ float sB[KPAD * NPAD];        //  13,312 B : dense spatial-resample matrix B_o[p', p]
    __shared__ float sC0[G_], sC1[G_];
    __shared__ int   sI0[G_], sI1[G_];

    const int blk    = blockIdx.x;           // 0..2047
    const int cinBlk = blk & 1;
    const int o      = (blk >> 1) & 7;
    const int cout   = blk >> 4;
    const int tid    = threadIdx.x;

    const float TWO_PI = 6.28318530717958647692f;

    // ---- phase 1: per-(o,i) circular-interp coefficients; zero sB ----
    if (tid < G_) {
        int i = tid;
        float q = in_H[i] - out_H[o];            // -out_H[o] + in_H[i]
        float m = fmodf(q, TWO_PI);
        if (m < 0.f) m += TWO_PI;                // jnp.mod semantics -> [0, 2pi)
        float pos  = m / (TWO_PI / G_);
        float base = floorf(pos);
        int   i0   = ((int)base) & 7;
        float t    = pos - base;
        sI0[i] = i0; sI1[i] = (i0 + 1) & 7;
        sC0[i] = 1.f - t; sC1[i] = t;
    }
    for (int idx = tid; idx < KPAD * NPAD; idx += THREADS) sB[idx] = 0.f;
    __syncthreads();

    // ---- phase 2: build B_o[p', p] = bilinear taps (rotated grid) * mask[p] ----
    if (tid < P_) {
        int p = tid;
        float ang = -out_H[o];
        float c = cosf(ang), s = sinf(ang);
        float x = grid_Rn[2 * p + 0];
        float y = grid_Rn[2 * p + 1];
        float rx = c * x - s * y;
        float ry = s * x + c * y;
        float px = (rx + 1.f) * 0.5f * (KS_ - 1);
        float py = (ry + 1.f) * 0.5f * (KS_ - 1);
        float x0 = floorf(px), y0 = floorf(py);
        float wx = px - x0,    wy = py - y0;
        float mk = mask[p];
        #pragma unroll
        for (int dy = 0; dy < 2; ++dy)
            #pragma unroll
            for (int dx = 0; dx < 2; ++dx) {
                float xi = x0 + (float)dx, yi = y0 + (float)dy;
                float wgt = (dx ? wx : 1.f - wx) * (dy ? wy : 1.f - wy);
                bool inb = (xi >= 0.f) && (xi <= (float)(KS_ - 1)) &&
                           (yi >= 0.f) && (yi <= (float)(KS_ - 1));
                if (inb) {
                    int psrc = (int)yi * KS_ + (int)xi;
                    sB[psrc * NPAD + p] += wgt * mk;   // distinct columns per thread: race-free
                }
            }
    }
    __syncthreads();

    // ---- phase 3: W1[r, p'] = c0*W[cout,cin,i0,p'] + c1*W[cout,cin,i1,p'] ----
    {
        int r   = tid;                       // one LDS row per thread
        int cin = cinBlk * CINBLK + (r >> 3);
        int i   = r & 7;
        const float* w0 = weight + (((size_t)cout * 128 + cin) * G_ + sI0[i]) * P_;
        const float* w1 = weight + (((size_t)cout * 128 + cin) * G_ + sI1[i]) * P_;
        float c0 = sC0[i], c1 = sC1[i];
        float* dst = sW1 + (size_t)r * W1STR;
        for (int p = 0; p < P_; ++p) dst[p] = c0 * w0[p] + c1 * w1[p];
        for (int p = P_; p < KPAD; ++p) dst[p] = 0.f;
    }
    __syncthreads();

    // ---- phase 4: fp32 WMMA: D[16x16] = A(16x4 of W1) x B(4x16 of B_o) + D ----
    const int wave = tid >> 5;    // 0..15
    const int lane = tid & 31;
    const int m16  = lane & 15;
    const int hi   = lane >> 4;   // 0: K pair {0,1} / M 0..7 ; 1: K pair {2,3} / M 8..15

    // Preload all B fragments into registers (shared across row tiles).
    // B 4x16 layout (mirror of documented A 16x4): vgpr j holds K rows (hi*2 + j).
    v2f bfrag[KSTEPS][4];
    #pragma unroll
    for (int k = 0; k < KSTEPS; ++k)
        #pragma unroll
        for (int n = 0; n < 4; ++n) {
            int col = n * 16 + m16;
            bfrag[k][n].x = sB[(k * 4 + hi * 2 + 0) * NPAD + col];
            bfrag[k][n].y = sB[(k * 4 + hi * 2 + 1) * NPAD + col];
        }

    // out idx = ((cout*8 + o)*1024 + cinBlk*512 + r) * 49 + p
    const size_t outBase = (((size_t)cout * G_ + o) * 1024 + (size_t)cinBlk * ROWS_) * P_;
    const v8f vzero = {0.f, 0.f, 0.f, 0.f, 0.f, 0.f, 0.f, 0.f};

    for (int rt = wave; rt < ROWS_ / 16; rt += 16) {   // 32 row tiles, 2 per wave
        v8f acc0 = vzero, acc1 = vzero, acc2 = vzero, acc3 = vzero;
        #pragma unroll
        for (int k = 0; k < KSTEPS; ++k) {
            // A 16x4: lane m16 holds row rt*16+m16, K columns (hi*2, hi*2+1)
            const float* arow = sW1 + (size_t)(rt * 16 + m16) * W1STR + k * 4 + hi * 2;
            v2f afrag; afrag.x = arow[0]; afrag.y = arow[1];
            acc0 = __builtin_amdgcn_wmma_f32_16x16x4_f32(false, afrag, false, bfrag[k][0],
                                                         (short)0, acc0, false, false);
            acc1 = __builtin_amdgcn_wmma_f32_16x16x4_f32(false, afrag, false, bfrag[k][1],
                                                         (short)0, acc1, false, false);
            acc2 = __builtin_amdgcn_wmma_f32_16x16x4_f32(false, afrag, false, bfrag[k][2],
                                                         (short)0, acc2, false, false);
            acc3 = __builtin_amdgcn_wmma_f32_16x16x4_f32(false, afrag, false, bfrag[k][3],
                                                         (short)0, acc3, false, false);
        }
        // D layout: vgpr j = row (hi*8 + j), column = n*16 + m16. Coalesced b32 stores.
        v8f accs[4] = {acc0, acc1, acc2, acc3};
        #pragma unroll
        for (int n = 0; n < 4; ++n) {
            int col = n * 16 + m16;
            if (col < P_) {
                #pragma unroll
                for (int j = 0; j < 8; ++j) {
                    int row = rt * 16 + hi * 8 + j;
                    out[outBase + (size_t)row * P_ + col] = accs[n][j];
                }
            }
        }
    }
}

extern "C" void kernel_launch(void* const* d_in, const int* in_sizes, int n_in,
                              void* d_out, int out_size, void* d_ws, size_t ws_size,
                              hipStream_t stream) {
    const float* weight  = (const float*)d_in[0];
    const float* in_H    = (const float*)d_in[1];
    const float* out_H   = (const float*)d_in[2];
    // d_in[3] = grid_H (identical to in_H/out_H grid, unused)
    const float* grid_Rn = (const float*)d_in[4];
    const float* mask    = (const float*)d_in[5];
    float* out = (float*)d_out;

    dim3 grid(128 * 8 * 2);   // cout * o * cinBlk
    dim3 block(THREADS);
    hipLaunchKernelGGL(gconv_expand_wmma, grid, block, 0, stream,
                       weight, in_H, out_H, grid_Rn, mask, out);
}